// PatchSampleLocalOneGroup_89189290868989
// MI455X (gfx1250) — compile-verified
//
#include <hip/hip_runtime.h>
#include <cstdint>
#include <cstddef>

// Problem constants from the reference (B, num_patches derived from in_sizes).
#define C_DIM   256
#define HW_DIM  256                 // H == W == 256
#define PW      4                   // PATCH_W
#define FEATS   (PW * PW * C_DIM)   // 4096 floats = 16 KB per (b, p)

// ---------------------------------------------------------------------------
// CDNA5 async global->LDS DMA (gfx1250, CDNA5 ISA).
// GVS mode: global address is the SGPR base plus a per-lane VGPR byte offset; the dword goes to the LDS address in vdst.
// Tracked by ASYNCcnt.
// ---------------------------------------------------------------------------
__device__ __forceinline__ void async_ld_b32_gvs(uint32_t lds_byte_addr,
                                                 uint32_t voff_bytes,
                                                 unsigned long long sbase) {
    asm volatile("global_load_async_to_lds_b32 %0, %1, %2"
                 :
                 : "v"(lds_byte_addr), "v"(voff_bytes), "s"(sbase)
                 : "memory");
}

__device__ __forceinline__ void wait_async_zero() {
    asm volatile("s_wait_asynccnt 0" ::: "memory");
}

// ---------------------------------------------------------------------------
// One workgroup per (batch, patch). 256 threads = 8 wave32.
// Phase 1: DMA-gather the 4x4xC patch into LDS (final flattened layout).
// Phase 2: deterministic block reduction of sum-of-squares.
// Phase 3: scale + coalesced float4 stores.
// ---------------------------------------------------------------------------
__global__ __launch_bounds__(256) void patch_sample_norm_kernel(
    const float* __restrict__ feat,        // (B, C, H, W)
    const int*   __restrict__ patch_ids,   // (num_patches, 2)
    float*       __restrict__ out,         // (B, num_patches, FEATS)
    int          num_patches)
{
    __shared__ __align__(16) float tile[FEATS];   // 16 KB staging tile
    __shared__ float wsum[8];                     // per-wave partial sums

    const int p = blockIdx.x;
    const int b = blockIdx.y;
    const int t = threadIdx.x;

    const int row0 = patch_ids[2 * p + 0];
    const int col0 = patch_ids[2 * p + 1];

    // Thread decomposition: t = ch_hi*16 + r*4 + c
    //  -> 4 consecutive lanes read 4 consecutive w (16 contiguous bytes).
    const int c    = t & 3;
    const int r    = (t >> 2) & 3;
    const int ch_h = t >> 4;               // 0..15, channel block of 16

    // Uniform SGPR base: feat + b*C*H*W  (stays in scalar registers).
    const unsigned long long sbase =
        (unsigned long long)(uintptr_t)(feat + (size_t)b * C_DIM * HW_DIM * HW_DIM);

    // Per-lane 32-bit byte offset of feat[b, ch_h*16, row0+r, col0+c] from sbase.
    // Max = (15*16*65536 + 255*256 + 255)*4 < 64 MB, fits signed 32-bit.
    uint32_t voff = (uint32_t)((((ch_h * 16) * HW_DIM * HW_DIM)
                                + (row0 + r) * HW_DIM + (col0 + c)) * 4);

    // LDS byte offset of tile[(r*4 + c)*C + ch_h*16]  (flattened output layout).
    const uint32_t lds_tile_base = (uint32_t)(uintptr_t)tile;  // low 32 bits = LDS offset
    uint32_t lds_elem = lds_tile_base
        + (uint32_t)((((r * PW + c) * C_DIM) + ch_h * 16) * 4);

    // Phase 1: 16 async b32 DMA loads per thread (ASYNCcnt peaks at 16 < 63).
    // Per-channel strides: +256 KB global, +4 B LDS.
    #pragma unroll
    for (int chl = 0; chl < 16; ++chl) {
        async_ld_b32_gvs(lds_elem + (uint32_t)(chl * 4),
                         voff + (uint32_t)(chl * HW_DIM * HW_DIM * 4),
                         sbase);
    }
    wait_async_zero();      // this wave's DMA complete (LDS writes visible)
    __syncthreads();        // all waves' DMA complete -> full tile visible

    // Phase 2: sum of squares. Thread t owns elements j*1024 + t*4 + {0..3}.
    float4 vals[4];
    float s = 0.0f;
    #pragma unroll
    for (int j = 0; j < 4; ++j) {
        vals[j] = *reinterpret_cast<const float4*>(&tile[j * 1024 + t * 4]);
        s += vals[j].x * vals[j].x + vals[j].y * vals[j].y
           + vals[j].z * vals[j].z + vals[j].w * vals[j].w;
    }
    // wave32 butterfly reduction (deterministic, warpSize == 32 on gfx1250)
    #pragma unroll
    for (int off = 16; off >= 1; off >>= 1)
        s += __shfl_xor(s, off, 32);
    if ((t & 31) == 0) wsum[t >> 5] = s;
    __syncthreads();

    float tot = 0.0f;
    #pragma unroll
    for (int i = 0; i < 8; ++i) tot += wsum[i];   // same order in every thread
    const float inv = 1.0f / (sqrtf(tot) + 1e-7f);

    // Phase 3: coalesced float4 stores (lanes write contiguous 512 B per wave).
    float* obase = out + ((size_t)b * num_patches + (size_t)p) * FEATS;
    #pragma unroll
    for (int j = 0; j < 4; ++j) {
        float4 v = vals[j];
        v.x *= inv; v.y *= inv; v.z *= inv; v.w *= inv;
        *reinterpret_cast<float4*>(&obase[j * 1024 + t * 4]) = v;
    }
}

// ---------------------------------------------------------------------------
// Harness entry point
// ---------------------------------------------------------------------------
extern "C" void kernel_launch(void* const* d_in, const int* in_sizes, int n_in,
                              void* d_out, int out_size, void* d_ws, size_t ws_size,
                              hipStream_t stream) {
    (void)n_in; (void)out_size; (void)d_ws; (void)ws_size;

    const float* feat      = (const float*)d_in[0];
    const int*   patch_ids = (const int*)d_in[1];
    float*       out       = (float*)d_out;

    // B from feat element count; num_patches from patch_ids element count.
    const int B           = in_sizes[0] / (C_DIM * HW_DIM * HW_DIM);
    const int num_patches = in_sizes[1] / 2;

    dim3 grid(num_patches, B);   // (256, 16)
    dim3 block(256);             // 8 wave32
    patch_sample_norm_kernel<<<grid, block, 0, stream>>>(feat, patch_ids, out,
                                                         num_patches);
}